// SparseSeparableConv_60979945669304
// MI455X (gfx1250) — compile-verified
//
#include <hip/hip_runtime.h>

typedef float v2f __attribute__((ext_vector_type(2)));
typedef float v4f __attribute__((ext_vector_type(4)));
typedef float v8f __attribute__((ext_vector_type(8)));

#define NPTS 262144
#define CIN 128
#define COUT 128
#define KOFF 9
#define TILE_M 64
#define DW_STRIDE 132   // padded row stride (floats) for dw tile: 528B rows, 16B aligned, conflict-free cols
#define WPW_STRIDE 130  // padded row stride (floats) for transposed w_pw: 8B aligned rows

__global__ __launch_bounds__(256)
void sparse_sepconv_fused(const float* __restrict__ feats,
                          const float* __restrict__ w_dw,
                          const float* __restrict__ w_pw,
                          const int* __restrict__ nidx,
                          const int* __restrict__ nmask,
                          float* __restrict__ out)
{
    __shared__ float s_dw[TILE_M * DW_STRIDE];    // 33792 B : depthwise result tile [m][k]
    __shared__ float s_wpwT[COUT * WPW_STRIDE];   // 66560 B : w_pw transposed [n][k]
    __shared__ float s_wdw[KOFF * CIN];           //  4608 B : depthwise weights

    const int t = threadIdx.x;
    const int block_row0 = blockIdx.x * TILE_M;

    // ---- cooperative preload: transpose w_pw into LDS; copy w_dw ----
    for (int i = t; i < CIN * COUT; i += 256) {
        int k = i >> 7;   // input channel (row of w_pw)
        int n = i & 127;  // output channel
        s_wpwT[n * WPW_STRIDE + k] = w_pw[i];
    }
    for (int i = t; i < KOFF * CIN; i += 256) s_wdw[i] = w_dw[i];
    __syncthreads();

    // ---- phase 1: depthwise gather + channelwise scale + sum over offsets ----
    {
        const int m  = t >> 2;        // point within tile: 0..63
        const int cq = (t & 3) * 32;  // this thread's 32-channel slice
        const int n_global = block_row0 + m;

        v4f acc[8];
        #pragma unroll
        for (int j = 0; j < 8; ++j) acc[j] = (v4f)0.0f;

        #pragma unroll
        for (int k = 0; k < KOFF; ++k) {
            const int   idx = nidx [k * NPTS + n_global];
            const float mk  = (float)nmask[k * NPTS + n_global];
            const float* frow = feats + (size_t)idx * CIN + cq;   // gather row (L2-resident)
            const float* wrow = s_wdw + k * CIN + cq;
            #pragma unroll
            for (int j = 0; j < 8; ++j) {
                v4f f = *(const v4f*)(frow + j * 4);
                v4f w = *(const v4f*)(wrow + j * 4);
                acc[j] += f * (w * mk);
            }
        }
        #pragma unroll
        for (int j = 0; j < 8; ++j)
            *(v4f*)&s_dw[m * DW_STRIDE + cq + j * 4] = acc[j];
    }
    __syncthreads();

    // ---- phase 2: pointwise GEMM via V_WMMA_F32_16X16X4_F32 ----
    {
        const int wave  = t >> 5;
        const int lane  = t & 31;
        const int l15   = lane & 15;
        const int khalf = (lane >> 4) << 1;  // lanes 0-15 -> K+{0,1}; lanes 16-31 -> K+{2,3}
        const int rbase = (wave >> 1) << 4;  // M band: 0,16,32,48
        const int cgrp  = (wave & 1) << 6;   // N half: 0 or 64

        v8f acc0 = (v8f)0.0f, acc1 = (v8f)0.0f, acc2 = (v8f)0.0f, acc3 = (v8f)0.0f;

        const float* arow = s_dw   + (rbase + l15) * DW_STRIDE + khalf;
        const float* b0r  = s_wpwT + (cgrp +  0 + l15) * WPW_STRIDE + khalf;
        const float* b1r  = s_wpwT + (cgrp + 16 + l15) * WPW_STRIDE + khalf;
        const float* b2r  = s_wpwT + (cgrp + 32 + l15) * WPW_STRIDE + khalf;
        const float* b3r  = s_wpwT + (cgrp + 48 + l15) * WPW_STRIDE + khalf;

        #pragma unroll 4
        for (int k0 = 0; k0 < CIN; k0 += 4) {
            v2f a  = *(const v2f*)(arow + k0);
            v2f b0 = *(const v2f*)(b0r  + k0);
            v2f b1 = *(const v2f*)(b1r  + k0);
            v2f b2 = *(const v2f*)(b2r  + k0);
            v2f b3 = *(const v2f*)(b3r  + k0);
            acc0 = __builtin_amdgcn_wmma_f32_16x16x4_f32(false, a, false, b0, (short)0, acc0, false, false);
            acc1 = __builtin_amdgcn_wmma_f32_16x16x4_f32(false, a, false, b1, (short)0, acc1, false, false);
            acc2 = __builtin_amdgcn_wmma_f32_16x16x4_f32(false, a, false, b2, (short)0, acc2, false, false);
            acc3 = __builtin_amdgcn_wmma_f32_16x16x4_f32(false, a, false, b3, (short)0, acc3, false, false);
        }

        // ---- store 4 16x16 f32 tiles (C/D layout: VGPR v = rows v and v+8) ----
        const int row_lo = block_row0 + rbase + ((lane >> 4) << 3);
        #pragma unroll
        for (int vg = 0; vg < 8; ++vg) {
            float* orow = out + (size_t)(row_lo + vg) * COUT + cgrp + l15;
            orow[0]  = acc0[vg];
            orow[16] = acc1[vg];
            orow[32] = acc2[vg];
            orow[48] = acc3[vg];
        }
    }
}

extern "C" void kernel_launch(void* const* d_in, const int* in_sizes, int n_in,
                              void* d_out, int out_size, void* d_ws, size_t ws_size,
                              hipStream_t stream) {
    const float* feats = (const float*)d_in[0];
    const float* w_dw  = (const float*)d_in[1];
    const float* w_pw  = (const float*)d_in[2];
    const int*   nidx  = (const int*)d_in[3];
    const int*   nmask = (const int*)d_in[4];
    float* out = (float*)d_out;

    dim3 grid(NPTS / TILE_M);  // 4096 blocks, 64 points each
    sparse_sepconv_fused<<<grid, 256, 0, stream>>>(feats, w_dw, w_pw, nidx, nmask, out);
}